// AGNNConv_6124623364170
// MI455X (gfx1250) — compile-verified
//
#include <hip/hip_runtime.h>

typedef __attribute__((ext_vector_type(2))) float v2f;
typedef __attribute__((ext_vector_type(8))) float v8f;

#define DFEAT 64

// Sign-aware float atomic-max via integer atomics (exact, deterministic).
// Works with buffer initialized to -inf; handles -0.0 via sign-bit test.
__device__ __forceinline__ void atomicMaxF32(float* addr, float val) {
    int ival = __float_as_int(val);
    if (ival >= 0) {
        atomicMax((int*)addr, ival);                       // non-negative floats order as ints
    } else {
        atomicMin((unsigned int*)addr, (unsigned int)ival); // negative floats reverse-order as uints
    }
}

__global__ void agnn_init(float* __restrict__ out, float* __restrict__ emax,
                          float* __restrict__ psum, int N, int ND) {
    int tid = blockIdx.x * blockDim.x + threadIdx.x;
    if (tid < ND) out[tid] = 0.0f;
    if (tid < N) { emax[tid] = -__builtin_inff(); psum[tid] = 0.0f; }
}

// One wave32 per node: 64 floats = float2 per lane, shuffle-reduce sum of squares.
__global__ void agnn_normalize(const float* __restrict__ feat,
                               float* __restrict__ norm_h, int N) {
    int wave = (blockIdx.x * blockDim.x + threadIdx.x) >> 5;
    int lane = threadIdx.x & 31;
    if (wave >= N) return;
    const float2* row = (const float2*)(feat + (long long)wave * DFEAT);
    float2 v = row[lane];
    float ss = v.x * v.x + v.y * v.y;
    #pragma unroll
    for (int o = 16; o > 0; o >>= 1) ss += __shfl_xor(ss, o, 32);
    float inv = 1.0f / fmaxf(sqrtf(ss), 1e-12f);
    float2 o2; o2.x = v.x * inv; o2.y = v.y * inv;
    ((float2*)(norm_h + (long long)wave * DFEAT))[lane] = o2;
}

// One wave per 16 edges: cosines = diag( A_src(16x64) * B_dst^T(64x16) )
// via 16 chained V_WMMA_F32_16X16X4_F32 (full f32 precision).
__global__ void agnn_edge_logits(const float* __restrict__ norm_h,
                                 const int* __restrict__ src,
                                 const int* __restrict__ dst,
                                 const float* __restrict__ beta,
                                 float* __restrict__ e_logits,
                                 float* __restrict__ emax, int E) {
    int wave = (blockIdx.x * blockDim.x + threadIdx.x) >> 5;
    int lane = threadIdx.x & 31;
    long long base = (long long)wave * 16;
    if (base >= E) return;            // wave-uniform exit: EXEC all-ones for WMMA
    int slot = lane & 15;
    int hi   = lane >> 4;
    long long eidx = base + slot;
    if (eidx >= E) eidx = E - 1;      // clamp for tail loads; writes guarded below
    int sRow = src[eidx];
    int dRow = dst[eidx];
    const float2* sp = (const float2*)(norm_h + (long long)sRow * DFEAT);
    const float2* dp = (const float2*)(norm_h + (long long)dRow * DFEAT);

    v8f acc = {};
    #pragma unroll
    for (int t = 0; t < 16; ++t) {
        // f32 A 16x4 layout: lanes 0-15 hold K=4t,4t+1; lanes 16-31 hold K=4t+2,4t+3
        float2 a2 = sp[2 * t + hi];
        float2 b2 = dp[2 * t + hi];
        v2f av; av[0] = a2.x; av[1] = a2.y;
        v2f bv; bv[0] = b2.x; bv[1] = b2.y;
        acc = __builtin_amdgcn_wmma_f32_16x16x4_f32(
            /*neg_a=*/false, av, /*neg_b=*/false, bv,
            /*c_mod=*/(short)0, acc, /*reuse_a=*/false, /*reuse_b=*/false);
    }

    // Extract diagonal: C[i][i] lives at (lane i, vgpr i) for i<8,
    // (lane 16+i, vgpr i-8) for i>=8.
    float myv = 0.0f;
    #pragma unroll
    for (int i = 0; i < 16; ++i) {
        float v = __shfl(acc[(i < 8) ? i : (i - 8)], (i < 8) ? i : (16 + i), 32);
        if (lane == i) myv = v;
    }

    if (hi == 0 && (base + slot) < E) {
        float ev = beta[0] * myv;
        e_logits[base + slot] = ev;
        atomicMaxF32(emax + dRow, ev);
    }
}

// One thread per edge: p = exp(e - emax[dst]); accumulate psum[dst].
__global__ void agnn_exp_psum(const float* __restrict__ emax,
                              const int* __restrict__ dst,
                              float* __restrict__ e_logits,
                              float* __restrict__ psum, int E) {
    int idx = blockIdx.x * blockDim.x + threadIdx.x;
    if (idx >= E) return;
    int d = dst[idx];
    float p = expf(e_logits[idx] - emax[d]);
    e_logits[idx] = p;
    atomicAdd(psum + d, p);
}

// One wave per edge: out[dst] += (p/psum[dst]) * feat[src]; float2 per lane.
__global__ void agnn_aggregate(const float* __restrict__ feat,
                               const int* __restrict__ src,
                               const int* __restrict__ dst,
                               const float* __restrict__ p_edge,
                               const float* __restrict__ psum,
                               float* __restrict__ out, int E) {
    int wave = (blockIdx.x * blockDim.x + threadIdx.x) >> 5;
    int lane = threadIdx.x & 31;
    if (wave >= E) return;
    int s = src[wave];
    int d = dst[wave];
    float w = p_edge[wave] / fmaxf(psum[d], 1e-38f);
    float2 f2 = ((const float2*)(feat + (long long)s * DFEAT))[lane];
    float* op = out + (long long)d * DFEAT + lane * 2;
    atomicAdd(op,     w * f2.x);
    atomicAdd(op + 1, w * f2.y);
}

extern "C" void kernel_launch(void* const* d_in, const int* in_sizes, int n_in,
                              void* d_out, int out_size, void* d_ws, size_t ws_size,
                              hipStream_t stream) {
    const float* feat = (const float*)d_in[0];
    const int*   src  = (const int*)d_in[1];
    const int*   dst  = (const int*)d_in[2];
    const float* beta = (const float*)d_in[3];

    const int D = DFEAT;
    const int N = in_sizes[0] / D;
    const int E = in_sizes[1];
    const int ND = N * D;

    float* out    = (float*)d_out;
    float* norm_h = (float*)d_ws;                       // N*D floats
    float* e_log  = norm_h + (size_t)N * D;             // E floats (logits, then p)
    float* emax   = e_log + (size_t)E;                  // N floats
    float* psum   = emax + (size_t)N;                   // N floats

    agnn_init<<<(ND + 255) / 256, 256, 0, stream>>>(out, emax, psum, N, ND);
    agnn_normalize<<<(N + 7) / 8, 256, 0, stream>>>(feat, norm_h, N);

    int groups = (E + 15) / 16;                          // 16 edges per wave
    agnn_edge_logits<<<(groups + 7) / 8, 256, 0, stream>>>(
        norm_h, src, dst, beta, e_log, emax, E);

    agnn_exp_psum<<<(E + 255) / 256, 256, 0, stream>>>(emax, dst, e_log, psum, E);

    agnn_aggregate<<<(E + 7) / 8, 256, 0, stream>>>(feat, src, dst, e_log, psum, out, E);
}